// DGLJTMPN_62199716381247
// MI455X (gfx1250) — compile-verified
//
#include <hip/hip_runtime.h>
#include <hip/hip_bf16.h>

typedef __attribute__((ext_vector_type(2))) float v2f;
typedef __attribute__((ext_vector_type(8))) float v8f;
typedef __attribute__((ext_vector_type(4))) unsigned int u32x4;
typedef __attribute__((ext_vector_type(8))) int i32x8;
typedef __attribute__((ext_vector_type(4))) int i32x4;

#define HDIM 256
#define DEPTH 4
#define LDSP 260   // padded LDS row stride (floats): banks = (4*row + k) % 64

__device__ __forceinline__ v8f wmma_f32_k4(v2f a, v2f b, v8f c) {
  // D = A(16x4) x B(4x16) + C(16x16), full fp32
  return __builtin_amdgcn_wmma_f32_16x16x4_f32(false, a, false, b, (short)0, c,
                                               false, false);
}

__device__ __forceinline__ unsigned as_lds_offset(void* p) {
  // generic -> LDS address space; integer value of AS(3) pointer is byte offset
  return (unsigned)(unsigned long long)(__attribute__((address_space(3))) char*)p;
}

// ---------------- elementwise / scatter helpers ----------------

__global__ void k_zero(float* __restrict__ p, long long n) {
  long long i = (long long)blockIdx.x * blockDim.x + threadIdx.x;
  long long st = (long long)gridDim.x * blockDim.x;
  for (; i < n; i += st) p[i] = 0.0f;
}

__global__ void k_copy(const float* __restrict__ src, float* __restrict__ dst,
                       long long n) {
  long long i = (long long)blockIdx.x * blockDim.x + threadIdx.x;
  long long st = (long long)gridDim.x * blockDim.x;
  for (; i < n; i += st) dst[i] = src[i];
}

// node_alpha[tgt[k]] += tree_m[teid[k]]   (total = K*H)
__global__ void k_scatter_alpha(const float* __restrict__ tree_m,
                                const int* __restrict__ tgt,
                                const int* __restrict__ teid,
                                float* __restrict__ node_alpha, long long total) {
  long long i = (long long)blockIdx.x * blockDim.x + threadIdx.x;
  long long st = (long long)gridDim.x * blockDim.x;
  for (; i < total; i += st) {
    long long k = i >> 8;
    int h = (int)(i & 255);
    atomicAdd(&node_alpha[(long long)tgt[k] * HDIM + h],
              tree_m[(long long)teid[k] * HDIM + h]);
  }
}

// tmp[e] = node_alpha[esrc[e]]   (total = E*H)
__global__ void k_tmp_init(const float* __restrict__ node_alpha,
                           const int* __restrict__ esrc,
                           float* __restrict__ tmp, long long total) {
  long long i = (long long)blockIdx.x * blockDim.x + threadIdx.x;
  long long st = (long long)gridDim.x * blockDim.x;
  for (; i < total; i += st) {
    long long e = i >> 8;
    int h = (int)(i & 255);
    tmp[i] = node_alpha[(long long)esrc[e] * HDIM + h];
  }
}

// tmp[lgd[l]] += msg[lgs[l]]   (total = L*H)
__global__ void k_lg_scatter(const float* __restrict__ msg,
                             const int* __restrict__ lgs,
                             const int* __restrict__ lgd,
                             float* __restrict__ tmp, long long total) {
  long long i = (long long)blockIdx.x * blockDim.x + threadIdx.x;
  long long st = (long long)gridDim.x * blockDim.x;
  for (; i < total; i += st) {
    long long l = i >> 8;
    int h = (int)(i & 255);
    atomicAdd(&tmp[(long long)lgd[l] * HDIM + h],
              msg[(long long)lgs[l] * HDIM + h]);
  }
}

// mplus[edst[e]] += msg[e]   (total = E*H)
__global__ void k_edge_scatter(const float* __restrict__ msg,
                               const int* __restrict__ edst,
                               float* __restrict__ mplus, long long total) {
  long long i = (long long)blockIdx.x * blockDim.x + threadIdx.x;
  long long st = (long long)gridDim.x * blockDim.x;
  for (; i < total; i += st) {
    long long e = i >> 8;
    int h = (int)(i & 255);
    atomicAdd(&mplus[(long long)edst[e] * HDIM + h],
              msg[(long long)e * HDIM + h]);
  }
}

// out[gid[n]] += hbuf[n]; cnt[gid[n]] += 1 (once per node)   (total = N*H)
__global__ void k_graph_accum(const float* __restrict__ hbuf,
                              const int* __restrict__ gid,
                              float* __restrict__ out, float* __restrict__ cnt,
                              long long total) {
  long long i = (long long)blockIdx.x * blockDim.x + threadIdx.x;
  long long st = (long long)gridDim.x * blockDim.x;
  for (; i < total; i += st) {
    long long n = i >> 8;
    int h = (int)(i & 255);
    int g = gid[n];
    atomicAdd(&out[(long long)g * HDIM + h], hbuf[i]);
    if (h == 0) atomicAdd(&cnt[g], 1.0f);
  }
}

__global__ void k_graph_div(float* __restrict__ out, const float* __restrict__ cnt,
                            long long total) {
  long long i = (long long)blockIdx.x * blockDim.x + threadIdx.x;
  long long st = (long long)gridDim.x * blockDim.x;
  for (; i < total; i += st) out[i] /= fmaxf(cnt[i >> 8], 1.0f);
}

// ---------------- WMMA GEMM kernels ----------------
// Block = 512 threads = 16 waves. One 16-row M-tile per block staged in LDS
// (bank-conflict-free padded stride), wave w owns N-tile w (16 cols).
// Per ISA layout: lane L holds A row L&15, K-pair 2*(L>>4); D VGPR i ->
// row mtile*16 + (L>>4)*8 + i, col ntile*16 + (L&15).

// msg_input = [x_nodes[esrc] | x_edges] @ Wi ; msg = relu(msg_input)
__global__ __launch_bounds__(512) void k_edge_init(
    const float* __restrict__ xn, const float* __restrict__ xe,
    const float* __restrict__ Wi, const int* __restrict__ esrc,
    float* __restrict__ msg_input, float* __restrict__ msg, int E) {
  __shared__ float As[16 * 42];   // stride 42: banks (42*row + k) % 64 distinct
  int base = blockIdx.x * 16;
  for (int i = threadIdx.x; i < 16 * 40; i += blockDim.x) {
    int rr = i / 40, kk = i - rr * 40;
    int e = base + rr;
    if (e >= E) e = E - 1;
    As[rr * 42 + kk] = (kk < 35) ? xn[(long long)esrc[e] * 35 + kk]
                                 : xe[(long long)e * 5 + (kk - 35)];
  }
  __syncthreads();
  int lane = threadIdx.x & 31;
  int ntile = threadIdx.x >> 5;
  int mrow = lane & 15;
  int khalf = lane >> 4;
  int col = ntile * 16 + mrow;
  v8f c = {0.f, 0.f, 0.f, 0.f, 0.f, 0.f, 0.f, 0.f};
  for (int k = 0; k < 40; k += 4) {
    int kk = k + 2 * khalf;
    v2f a = *(const v2f*)&As[mrow * 42 + kk];          // ds_load_b64
    v2f b;
    b.x = Wi[kk * HDIM + col];
    b.y = Wi[(kk + 1) * HDIM + col];
    c = wmma_f32_k4(a, b, c);
  }
  int mbase = base + khalf * 8;
  long long rowoff = (long long)mbase * HDIM + col;
  if (base + 16 <= E) {          // uniform fast path (E%16==0 in practice)
#pragma unroll
    for (int i = 0; i < 8; i++) {
      long long idx = rowoff + (long long)i * HDIM;
      float v = c[i];
      msg_input[idx] = v;
      msg[idx] = fmaxf(v, 0.0f);
    }
  } else {
    for (int i = 0; i < 8; i++) {
      if (mbase + i < E) {
        long long idx = rowoff + (long long)i * HDIM;
        float v = c[i];
        msg_input[idx] = v;
        msg[idx] = fmaxf(v, 0.0f);
      }
    }
  }
}

// msg = relu(msg_input + A @ Wh)   A = tmp[E,256]
// A-tile (16x256 f32 = 16KB) staged into LDS by the Tensor Data Mover with
// hardware row padding (pad_interval=7 -> every 256 DWORDs, pad_amount=3 ->
// +4 DWORDs) giving the conflict-free 260-float LDS stride.
__global__ __launch_bounds__(512) void k_gemm_wh(
    const float* __restrict__ A, const float* __restrict__ Wh,
    const float* __restrict__ msg_input, float* __restrict__ msg, int E) {
  __shared__ float As[16 * LDSP];
  int base = blockIdx.x * 16;
  if (threadIdx.x < 32) {        // wave 0 issues the TDM DMA (EXEC-independent)
    unsigned long long gaddr =
        (unsigned long long)(const void*)(A + (long long)base * HDIM);
    unsigned dim1 = (unsigned)(E - base);   // OOB rows read as zero
    u32x4 g0;
    g0[0] = 1u;                              // count=1 (valid user descriptor)
    g0[1] = as_lds_offset((void*)As);        // lds_addr
    g0[2] = (unsigned)gaddr;                 // global_addr[31:0]
    g0[3] = (unsigned)((gaddr >> 32) & 0x01FFFFFFu) | (2u << 30);  // [56:32]|type=2
    i32x8 g1;
    g1[0] = (int)((2u << 16)        // data_size = 4B
                  | (1u << 20)      // pad_enable
                  | (7u << 22)      // pad_interval: 256 DWORDs
                  | (3u << 25));    // pad_amount: 4 DWORDs
    g1[1] = (int)(256u << 16);                              // tensor_dim0[15:0]
    g1[2] = (int)((dim1 & 0xFFFFu) << 16);                  // dim0[31:16]=0 | dim1[15:0]
    g1[3] = (int)(((dim1 >> 16) & 0xFFFFu) | (256u << 16)); // dim1[31:16] | tile_dim0=256
    g1[4] = 16;                                             // tile_dim1=16, tile_dim2=0
    g1[5] = 256;                                            // tensor_dim0_stride[31:0]
    g1[6] = 0;                                              // stride hi | dim1_stride lo
    g1[7] = 0;
    i32x4 gz4 = {0, 0, 0, 0};
    i32x8 gz8 = {0, 0, 0, 0, 0, 0, 0, 0};
    __builtin_amdgcn_tensor_load_to_lds(g0, g1, gz4, gz4, gz8, 0);
    __builtin_amdgcn_s_wait_tensorcnt(0);
  }
  __syncthreads();
  int lane = threadIdx.x & 31;
  int ntile = threadIdx.x >> 5;
  int mrow = lane & 15;
  int khalf = lane >> 4;
  int col = ntile * 16 + mrow;
  v8f c = {0.f, 0.f, 0.f, 0.f, 0.f, 0.f, 0.f, 0.f};
  for (int k = 0; k < HDIM; k += 4) {
    int kk = k + 2 * khalf;
    v2f a = *(const v2f*)&As[mrow * LDSP + kk];
    v2f b;
    b.x = Wh[kk * HDIM + col];
    b.y = Wh[(kk + 1) * HDIM + col];
    c = wmma_f32_k4(a, b, c);
  }
  int mbase = base + khalf * 8;
  long long rowoff = (long long)mbase * HDIM + col;
  if (base + 16 <= E) {
#pragma unroll
    for (int i = 0; i < 8; i++) {
      long long idx = rowoff + (long long)i * HDIM;
      msg[idx] = fmaxf(msg_input[idx] + c[i], 0.0f);
    }
  } else {
    for (int i = 0; i < 8; i++) {
      if (mbase + i < E) {
        long long idx = rowoff + (long long)i * HDIM;
        msg[idx] = fmaxf(msg_input[idx] + c[i], 0.0f);
      }
    }
  }
}

// hbuf = relu([x_nodes | mplus] @ Wo + bo)   K = 291, padded to 292
__global__ __launch_bounds__(512) void k_gemm_out(
    const float* __restrict__ xn, const float* __restrict__ mplus,
    const float* __restrict__ Wo, const float* __restrict__ bo,
    float* __restrict__ hbuf, int N) {
  __shared__ float As[16 * 292];  // stride 292: (36*row + k) % 64 all distinct
  int base = blockIdx.x * 16;
  for (int i = threadIdx.x; i < 16 * 292; i += blockDim.x) {
    int rr = i / 292, kk = i - rr * 292;
    int node = base + rr;
    if (node >= N) node = N - 1;
    float v = 0.0f;
    if (kk < 35) v = xn[(long long)node * 35 + kk];
    else if (kk < 291) v = mplus[(long long)node * HDIM + (kk - 35)];
    As[i] = v;
  }
  __syncthreads();
  int lane = threadIdx.x & 31;
  int ntile = threadIdx.x >> 5;
  int mrow = lane & 15;
  int khalf = lane >> 4;
  int col = ntile * 16 + mrow;
  v8f c = {0.f, 0.f, 0.f, 0.f, 0.f, 0.f, 0.f, 0.f};
  for (int k = 0; k < 292; k += 4) {
    int kk = k + 2 * khalf;
    v2f a = *(const v2f*)&As[mrow * 292 + kk];
    v2f b;
    b.x = (kk < 291) ? Wo[(long long)kk * HDIM + col] : 0.0f;
    b.y = (kk + 1 < 291) ? Wo[(long long)(kk + 1) * HDIM + col] : 0.0f;
    c = wmma_f32_k4(a, b, c);
  }
  int mbase = base + khalf * 8;
  long long rowoff = (long long)mbase * HDIM + col;
  if (base + 16 <= N) {
#pragma unroll
    for (int i = 0; i < 8; i++) {
      long long idx = rowoff + (long long)i * HDIM;
      hbuf[idx] = fmaxf(c[i] + bo[col], 0.0f);
    }
  } else {
    for (int i = 0; i < 8; i++) {
      if (mbase + i < N) {
        long long idx = rowoff + (long long)i * HDIM;
        hbuf[idx] = fmaxf(c[i] + bo[col], 0.0f);
      }
    }
  }
}

// ---------------- launcher ----------------

extern "C" void kernel_launch(void* const* d_in, const int* in_sizes, int n_in,
                              void* d_out, int out_size, void* d_ws, size_t ws_size,
                              hipStream_t stream) {
  (void)n_in; (void)ws_size;
  const float* xn     = (const float*)d_in[0];   // [N,35]
  const float* xe     = (const float*)d_in[1];   // [E,5]
  const float* tree_m = (const float*)d_in[2];   // [Et,H]
  const float* Wi     = (const float*)d_in[3];   // [40,H]
  const float* Wh     = (const float*)d_in[4];   // [H,H]
  const float* Wo     = (const float*)d_in[5];   // [291,H]
  const float* bo     = (const float*)d_in[6];   // [H]
  const int* esrc = (const int*)d_in[7];
  const int* edst = (const int*)d_in[8];
  const int* lgs  = (const int*)d_in[9];
  const int* lgd  = (const int*)d_in[10];
  const int* tgt  = (const int*)d_in[11];
  const int* teid = (const int*)d_in[12];
  const int* gid  = (const int*)d_in[13];

  const int N = in_sizes[13];
  const int E = in_sizes[7];
  const int L = in_sizes[9];
  const int K = in_sizes[11];
  const int G = out_size / HDIM;

  float* out = (float*)d_out;
  float* msg_input  = (float*)d_ws;
  float* msg        = msg_input + (long long)E * HDIM;
  float* tmp        = msg + (long long)E * HDIM;
  float* node_alpha = tmp + (long long)E * HDIM;
  float* cnt        = node_alpha + (long long)N * HDIM;
  float* mplus      = tmp;                              // reuse after BP loop
  float* hbuf       = tmp + (long long)N * HDIM;        // 2*N*H < E*H

  const int TB = 256;

  // tree-message alignment: node_alpha
  k_zero<<<2048, TB, 0, stream>>>(node_alpha, (long long)N * HDIM);
  k_scatter_alpha<<<2048, TB, 0, stream>>>(tree_m, tgt, teid, node_alpha,
                                           (long long)K * HDIM);

  // msg_input / msg
  int mtE = (E + 15) / 16;
  k_edge_init<<<mtE, 512, 0, stream>>>(xn, xe, Wi, esrc, msg_input, msg, E);

  // loopy BP over the line graph
  for (int it = 0; it < DEPTH - 1; ++it) {
    k_tmp_init<<<4096, TB, 0, stream>>>(node_alpha, esrc, tmp,
                                        (long long)E * HDIM);
    k_lg_scatter<<<4096, TB, 0, stream>>>(msg, lgs, lgd, tmp,
                                          (long long)L * HDIM);
    k_gemm_wh<<<mtE, 512, 0, stream>>>(tmp, Wh, msg_input, msg, E);
  }

  // node readout: mplus = node_alpha + segment_sum(msg, edge_dst)
  k_copy<<<2048, TB, 0, stream>>>(node_alpha, mplus, (long long)N * HDIM);
  k_edge_scatter<<<4096, TB, 0, stream>>>(msg, edst, mplus, (long long)E * HDIM);

  int mtN = (N + 15) / 16;
  k_gemm_out<<<mtN, 512, 0, stream>>>(xn, mplus, Wo, bo, hbuf, N);

  // per-graph mean
  k_zero<<<256, TB, 0, stream>>>(out, (long long)G * HDIM);
  k_zero<<<64, TB, 0, stream>>>(cnt, (long long)G);
  k_graph_accum<<<2048, TB, 0, stream>>>(hbuf, gid, out, cnt,
                                         (long long)N * HDIM);
  k_graph_div<<<256, TB, 0, stream>>>(out, cnt, (long long)G * HDIM);
}